// FuncConv_30073361006825
// MI455X (gfx1250) — compile-verified
//
#include <hip/hip_runtime.h>

#define N_TOT   131072
#define LVL     8
#define NPL     16384
#define DEG     16
#define DIM     128
#define HID     64
#define NEGS    0.01f

typedef float v2f __attribute__((ext_vector_type(2)));
typedef float v8f __attribute__((ext_vector_type(8)));

// D = A(16x4) * B(4x16) + C  in full fp32 (matches f32 reference numerics)
#define WMMA_F32(a, b, c) \
  __builtin_amdgcn_wmma_f32_16x16x4_f32(false, (a), false, (b), (short)0, (c), false, false)

// A fragment from LDS: lane holds row m0+lc, k-pair selected by lane half
__device__ __forceinline__ v2f lds_a(const float* S, int ld, int m0, int k0, int half, int lc) {
  const float* p = S + (m0 + lc) * ld + k0 + 2 * half;
  v2f a; a.x = p[0]; a.y = p[1]; return a;
}
// B fragment from global weights W[K][ld]: lane holds col n0+lc, k-pair by half
__device__ __forceinline__ v2f glb_b(const float* W, int ld, int n0, int k0, int half, int lc) {
  const float* p = W + (k0 + 2 * half) * ld + n0 + lc;
  v2f b; b.x = p[0]; b.y = p[ld]; return b;
}
__device__ __forceinline__ float lrelu(float x) { return x > 0.f ? x : x * NEGS; }

__global__ __launch_bounds__(256)
void dag_level_kernel(float* __restrict__ h,
                      const int* __restrict__ esrc, const int* __restrict__ emask,
                      const float* __restrict__ iw1, const float* __restrict__ ib1,
                      const float* __restrict__ iw2, const float* __restrict__ ib2,
                      const float* __restrict__ iw3, const float* __restrict__ ib3,
                      const float* __restrict__ aw1, const float* __restrict__ ab1,
                      const float* __restrict__ aw2, const float* __restrict__ ab2,
                      const float* __restrict__ aw3, const float* __restrict__ ab3,
                      int off)
{
  __shared__ float    s_msg[64 * 132];   // 64 message rows, padded stride
  __shared__ float    s_act[64 * 68];    // activations (reused layer1->2->3 and and-MLP)
  __shared__ float    s_ngh[8 * 132];    // aggregated neighbor features, 8 dst rows
  __shared__ unsigned s_mask[8];         // 16 edge-mask bits per dst node

  const int t    = threadIdx.x;
  const int w    = t >> 5;       // wave id 0..7
  const int lane = t & 31;
  const int half = lane >> 4;    // lane half (WMMA K/M split)
  const int lc   = lane & 15;    // row (A) / col (B,C) within tile
  const int dst0 = blockIdx.x * 8;

  if (t < 8) s_mask[t] = 0u;
  __syncthreads();
  if (t < 128) {
    int dl = t >> 4, e = t & 15;
    if (emask[(dst0 + dl) * DEG + e] != 0) atomicOr(&s_mask[dl], 1u << e);
  }

  for (int chunk = 0; chunk < 2; ++chunk) {
    // ---- gather 64 message rows (4 dst nodes x 16 edges, dst-major) ----
    for (int p = 0; p < 8; ++p) {
      int r  = p * 8 + (t >> 5);          // local row 0..63
      int dl = chunk * 4 + (r >> 4);
      int e  = r & 15;
      int src = esrc[(dst0 + dl) * DEG + e];
      int c4  = t & 31;                   // float4 column chunk
      float4 v = *(const float4*)(h + (size_t)src * DIM + c4 * 4);
      *(float4*)(&s_msg[r * 132 + c4 * 4]) = v;
    }
    __syncthreads();

    // ---- inv layer 1: [64x128] @ [128x64], bias + leaky ----
    {
      int n0  = (w & 3) * 16;
      int m0a = (w >> 2) * 32;            // two m-tiles per wave share B fragment
      v8f c0 = {}; v8f c1 = {};
      for (int k0 = 0; k0 < DIM; k0 += 4) {
        v2f b  = glb_b(iw1, HID, n0, k0, half, lc);
        v2f a0 = lds_a(s_msg, 132, m0a,      k0, half, lc);
        v2f a1 = lds_a(s_msg, 132, m0a + 16, k0, half, lc);
        c0 = WMMA_F32(a0, b, c0);
        c1 = WMMA_F32(a1, b, c1);
      }
      float bia = ib1[n0 + lc];
      #pragma unroll
      for (int v = 0; v < 8; ++v) {
        s_act[(m0a      + v + 8 * half) * 68 + n0 + lc] = lrelu(c0[v] + bia);
        s_act[(m0a + 16 + v + 8 * half) * 68 + n0 + lc] = lrelu(c1[v] + bia);
      }
    }
    __syncthreads();

    // ---- inv layer 2: [64x64] @ [64x64], in-place on s_act ----
    {
      int n0  = (w & 3) * 16;
      int m0a = (w >> 2) * 32;
      v8f c0 = {}; v8f c1 = {};
      for (int k0 = 0; k0 < HID; k0 += 4) {
        v2f b  = glb_b(iw2, HID, n0, k0, half, lc);
        v2f a0 = lds_a(s_act, 68, m0a,      k0, half, lc);
        v2f a1 = lds_a(s_act, 68, m0a + 16, k0, half, lc);
        c0 = WMMA_F32(a0, b, c0);
        c1 = WMMA_F32(a1, b, c1);
      }
      __syncthreads();   // all reads of s_act complete before overwrite
      float bia = ib2[n0 + lc];
      #pragma unroll
      for (int v = 0; v < 8; ++v) {
        s_act[(m0a      + v + 8 * half) * 68 + n0 + lc] = lrelu(c0[v] + bia);
        s_act[(m0a + 16 + v + 8 * half) * 68 + n0 + lc] = lrelu(c1[v] + bia);
      }
    }
    __syncthreads();

    // ---- inv layer 3 + mask select + mean over 16 edges (in-fragment) ----
    {
      int n0 = w * 16;                    // wave owns one 16-col strip, all 4 m-tiles
      v8f c0 = {}; v8f c1 = {}; v8f c2 = {}; v8f c3 = {};
      for (int k0 = 0; k0 < HID; k0 += 4) {
        v2f b  = glb_b(iw3, DIM, n0, k0, half, lc);
        v2f a0 = lds_a(s_act, 68,  0, k0, half, lc);
        v2f a1 = lds_a(s_act, 68, 16, k0, half, lc);
        v2f a2 = lds_a(s_act, 68, 32, k0, half, lc);
        v2f a3 = lds_a(s_act, 68, 48, k0, half, lc);
        c0 = WMMA_F32(a0, b, c0);
        c1 = WMMA_F32(a1, b, c1);
        c2 = WMMA_F32(a2, b, c2);
        c3 = WMMA_F32(a3, b, c3);
      }
      float b3 = ib3[n0 + lc];
      v8f cc[4] = { c0, c1, c2, c3 };
      #pragma unroll
      for (int mt = 0; mt < 4; ++mt) {    // each m-tile = one dst node's 16 edges
        int dl = chunk * 4 + mt;
        unsigned mb = s_mask[dl];
        float sum = 0.f;
        #pragma unroll
        for (int v = 0; v < 8; ++v) {
          int e = v + 8 * half;           // edge index == row within tile
          float inv = cc[mt][v] + b3;
          float msg = s_msg[(mt * 16 + e) * 132 + n0 + lc];
          sum += ((mb >> e) & 1u) ? inv : msg;
        }
        sum += __shfl_xor(sum, 16, 32);   // combine rows 0-7 with rows 8-15
        if (half == 0) s_ngh[dl * 132 + n0 + lc] = sum * (1.0f / 16.0f);
      }
    }
    __syncthreads();
  }

  // ---- and-MLP on 8 aggregated rows (padded to one 16-row tile) ----
  // layer 1: [16x128] @ [128x64]
  {
    v8f c = {};
    if (w < 4) {
      int n0 = w * 16;
      for (int k0 = 0; k0 < DIM; k0 += 4) {
        v2f b = glb_b(aw1, HID, n0, k0, half, lc);
        v2f a;
        if (lc < 8) { a.x = s_ngh[lc * 132 + k0 + 2 * half];
                      a.y = s_ngh[lc * 132 + k0 + 2 * half + 1]; }
        else        { a.x = 0.f; a.y = 0.f; }
        c = WMMA_F32(a, b, c);
      }
      float bia = ab1[n0 + lc];
      #pragma unroll
      for (int v = 0; v < 8; ++v)
        s_act[(v + 8 * half) * 68 + n0 + lc] = lrelu(c[v] + bia);
    }
  }
  __syncthreads();
  // layer 2: [16x64] @ [64x64], in-place
  {
    v8f c = {};
    if (w < 4) {
      int n0 = w * 16;
      for (int k0 = 0; k0 < HID; k0 += 4) {
        v2f b = glb_b(aw2, HID, n0, k0, half, lc);
        v2f a = lds_a(s_act, 68, 0, k0, half, lc);
        c = WMMA_F32(a, b, c);
      }
    }
    __syncthreads();
    if (w < 4) {
      int n0 = w * 16;
      float bia = ab2[n0 + lc];
      #pragma unroll
      for (int v = 0; v < 8; ++v)
        s_act[(v + 8 * half) * 68 + n0 + lc] = lrelu(c[v] + bia);
    }
  }
  __syncthreads();
  // layer 3: [16x64] @ [64x128] -> h[off + dst]
  {
    int n0 = w * 16;
    v8f c = {};
    for (int k0 = 0; k0 < HID; k0 += 4) {
      v2f b = glb_b(aw3, DIM, n0, k0, half, lc);
      v2f a = lds_a(s_act, 68, 0, k0, half, lc);
      c = WMMA_F32(a, b, c);
    }
    float b3 = ab3[n0 + lc];
    if (half == 0) {                      // rows 0..7 are the 8 valid dst rows
      #pragma unroll
      for (int v = 0; v < 8; ++v)
        h[(size_t)(off + dst0 + v) * DIM + n0 + lc] = c[v] + b3;
    }
  }
}

extern "C" void kernel_launch(void* const* d_in, const int* in_sizes, int n_in,
                              void* d_out, int out_size, void* d_ws, size_t ws_size,
                              hipStream_t stream) {
  const float* h_in  = (const float*)d_in[0];
  const int*   esrc  = (const int*)d_in[1];
  const int*   emask = (const int*)d_in[2];
  const float* iw1 = (const float*)d_in[3];
  const float* ib1 = (const float*)d_in[4];
  const float* iw2 = (const float*)d_in[5];
  const float* ib2 = (const float*)d_in[6];
  const float* iw3 = (const float*)d_in[7];
  const float* ib3 = (const float*)d_in[8];
  const float* aw1 = (const float*)d_in[9];
  const float* ab1 = (const float*)d_in[10];
  const float* aw2 = (const float*)d_in[11];
  const float* ab2 = (const float*)d_in[12];
  const float* aw3 = (const float*)d_in[13];
  const float* ab3 = (const float*)d_in[14];
  float* out = (float*)d_out;

  // h evolves in place inside d_out; level 0 rows stay untouched.
  hipMemcpyAsync(out, h_in, (size_t)N_TOT * DIM * sizeof(float),
                 hipMemcpyDeviceToDevice, stream);

  for (int lev = 1; lev < LVL; ++lev) {
    dag_level_kernel<<<NPL / 8, 256, 0, stream>>>(
        out,
        esrc  + (size_t)(lev - 1) * NPL * DEG,
        emask + (size_t)(lev - 1) * NPL * DEG,
        iw1, ib1, iw2, ib2, iw3, ib3,
        aw1, ab1, aw2, ab2, aw3, ab3,
        lev * NPL);
  }
}